// POYOPlusE_12859132084759
// MI455X (gfx1250) — compile-verified
//
#include <hip/hip_runtime.h>
#include <hip/hip_bf16.h>
#include <math.h>
#include <stdint.h>

// ---------------- model dims ----------------
#define DIMM   512
#define HD     64
#define NL     64
#define BB     16
#define TIN    32768
#define TOUT   8192
#define BNL    (BB*NL)      // 1024
#define SH     8
#define FF     2048         // reference FF = 4*DIM = 2048
#define TWOFF  4096

typedef float v2f __attribute__((ext_vector_type(2)));
typedef float v8f __attribute__((ext_vector_type(8)));

// ---------------- wave32 helpers ----------------
__device__ __forceinline__ float wred_sum(float v) {
  for (int m = 16; m > 0; m >>= 1) v += __shfl_xor(v, m, 32);
  return v;
}
__device__ __forceinline__ float wred_max(float v) {
  for (int m = 16; m > 0; m >>= 1) v = fmaxf(v, __shfl_xor(v, m, 32));
  return v;
}

// ---------------- CDNA5 async global->LDS (ASYNCcnt-tracked DMA into LDS) ----------------
// Flat->LDS mapping truncates upper bits (ISA 10.2): low 32 bits of a generic
// pointer to __shared__ data are the LDS byte offset used by VDST.
__device__ __forceinline__ unsigned lds_off_of(const void* p) {
  return (unsigned)(uintptr_t)p;
}
__device__ __forceinline__ void async_b128(unsigned lds, const void* g) {
  asm volatile("global_load_async_to_lds_b128 %0, %1, off"
               :: "v"(lds), "v"((unsigned long long)(uintptr_t)g)
               : "memory");
}
__device__ __forceinline__ void wait_async0() {
  asm volatile("s_wait_asynccnt 0" ::: "memory");
}

// ---------------- batch range offsets (ragged masks are contiguous) ----------------
__global__ __launch_bounds__(256) void k_offsets(const int* __restrict__ ib,
                                                 const int* __restrict__ ob,
                                                 int* __restrict__ offs) {
  __shared__ int cin[BB], cout[BB];
  int tid = threadIdx.x;
  if (tid < BB) { cin[tid] = 0; cout[tid] = 0; }
  __syncthreads();
  for (int i = tid; i < TIN; i += 256) atomicAdd(&cin[ib[i]], 1);
  for (int i = tid; i < TOUT; i += 256) atomicAdd(&cout[ob[i]], 1);
  __syncthreads();
  if (tid == 0) {
    int a = 0, b = 0;
    for (int k = 0; k < BB; k++) {
      offs[k] = a;      a += cin[k];
      offs[17 + k] = b; b += cout[k];
      offs[34 + k] = k * NL;
    }
    offs[16] = a; offs[33] = b; offs[50] = BB * NL;
  }
}

// ---------------- rotary cos/sin tables: th[t,d] = t * 10000^(-(d%32)/32) ----------------
__global__ __launch_bounds__(256) void k_rope_tables(const float* __restrict__ ts,
                                                     float* __restrict__ cosT,
                                                     float* __restrict__ sinT, int T) {
  int i = blockIdx.x * 256 + threadIdx.x;
  if (i >= T * 32) return;
  int t = i >> 5, j = i & 31;
  float inv = expf(-(float)j * (9.210340371976184f / 32.0f)); // 10000^(-j/32)
  float a = ts[t] * inv;
  float c = cosf(a), s = sinf(a);
  cosT[t * HD + j] = c; cosT[t * HD + j + 32] = c;
  sinT[t * HD + j] = s; sinT[t * HD + j + 32] = s;
}

// ---------------- fused embed(unit+tok) + LayerNorm (wave per row) ----------------
__global__ __launch_bounds__(256) void k_embed_ln(const int* __restrict__ i1,
                                                  const int* __restrict__ i2,
                                                  const float* __restrict__ e1,
                                                  const float* __restrict__ e2,
                                                  const float* __restrict__ g,
                                                  const float* __restrict__ b,
                                                  float* __restrict__ Y, int T) {
  int row = blockIdx.x * 8 + (threadIdx.x >> 5);
  if (row >= T) return;
  int lane = threadIdx.x & 31;
  const float* a1 = e1 + (size_t)i1[row] * DIMM;
  const float* a2 = e2 + (size_t)i2[row] * DIMM;
  float vals[16], s = 0.f, s2 = 0.f;
  #pragma unroll
  for (int i = 0; i < 16; i++) {
    float v = a1[lane + 32 * i] + a2[lane + 32 * i];
    vals[i] = v; s += v; s2 += v * v;
  }
  s = wred_sum(s); s2 = wred_sum(s2);
  float mean = s * (1.f / DIMM);
  float var = s2 * (1.f / DIMM) - mean * mean;
  float r = rsqrtf(var + 1e-5f);
  float* y = Y + (size_t)row * DIMM;
  #pragma unroll
  for (int i = 0; i < 16; i++) {
    int d = lane + 32 * i;
    y[d] = (vals[i] - mean) * r * g[d] + b[d];
  }
}

// ---------------- plain gathers ----------------
__global__ __launch_bounds__(256) void k_embed1(const int* __restrict__ i1,
                                                const float* __restrict__ e1,
                                                float* __restrict__ Y, int total) {
  int i = blockIdx.x * 256 + threadIdx.x;
  if (i >= total) return;
  int t = i >> 9, d = i & 511;
  Y[i] = e1[(size_t)i1[t] * DIMM + d];
}
__global__ __launch_bounds__(256) void k_embed2(const int* __restrict__ i1,
                                                const int* __restrict__ i2,
                                                const float* __restrict__ e1,
                                                const float* __restrict__ e2,
                                                float* __restrict__ Y, int total) {
  int i = blockIdx.x * 256 + threadIdx.x;
  if (i >= total) return;
  int t = i >> 9, d = i & 511;
  Y[i] = e1[(size_t)i1[t] * DIMM + d] + e2[(size_t)i2[t] * DIMM + d];
}

// ---------------- LayerNorm (wave per row) ----------------
__global__ __launch_bounds__(256) void k_ln(const float* __restrict__ X,
                                            const float* __restrict__ g,
                                            const float* __restrict__ b,
                                            float* __restrict__ Y, int T) {
  int row = blockIdx.x * 8 + (threadIdx.x >> 5);
  if (row >= T) return;
  int lane = threadIdx.x & 31;
  const float* x = X + (size_t)row * DIMM;
  float vals[16], s = 0.f, s2 = 0.f;
  #pragma unroll
  for (int i = 0; i < 16; i++) { float v = x[lane + 32 * i]; vals[i] = v; s += v; s2 += v * v; }
  s = wred_sum(s); s2 = wred_sum(s2);
  float mean = s * (1.f / DIMM);
  float var = s2 * (1.f / DIMM) - mean * mean;
  float r = rsqrtf(var + 1e-5f);
  float* y = Y + (size_t)row * DIMM;
  #pragma unroll
  for (int i = 0; i < 16; i++) { int d = lane + 32 * i; y[d] = (vals[i] - mean) * r * g[d] + b[d]; }
}

// ================= WMMA fp32 GEMM: C[M,N] = A[M,K] @ W[K,N] (+bias)(+res) =================
// 64x64 tile / workgroup, 8 waves, each wave -> 16x32 via two v_wmma_f32_16x16x4_f32 chains.
// K-chunks of 16 double-buffered in LDS via async global->LDS B128 DMA (ASYNCcnt).
__global__ __launch_bounds__(256) void k_gemm(const float* __restrict__ A,
                                              const float* __restrict__ W,
                                              const float* __restrict__ bias,
                                              const float* __restrict__ res,
                                              float* __restrict__ C,
                                              int M, int N, int K) {
  __shared__ __align__(16) float As[2][64][20];   // row stride 80B (16B aligned)
  __shared__ __align__(16) float Bs[2][16][68];   // row stride 272B (16B aligned)
  int ntiles = N >> 6;
  int m0 = (blockIdx.x / ntiles) * 64;
  int n0 = (blockIdx.x % ntiles) * 64;
  int tid = threadIdx.x, wave = tid >> 5, lane = tid & 31;
  int mt = (wave >> 1) * 16;       // 0,16,32,48
  int nt = (wave & 1) * 32;        // 0 or 32
  int krow = (lane >> 4) << 1;     // 0 or 2 (A 16x4 f32 lane layout)
  int ml = lane & 15;
  int ar = tid >> 2, ac = (tid & 3) * 4;     // A staging: one B128 per thread
  int br = tid >> 4, bc = (tid & 15) * 4;    // B staging: one B128 per thread
  v8f c0 = {0.f,0.f,0.f,0.f,0.f,0.f,0.f,0.f};
  v8f c1 = {0.f,0.f,0.f,0.f,0.f,0.f,0.f,0.f};

  auto issue = [&](int kc, int buf) {
    async_b128(lds_off_of(&As[buf][ar][ac]), A + (size_t)(m0 + ar) * K + kc + ac);
    async_b128(lds_off_of(&Bs[buf][br][bc]), W + (size_t)(kc + br) * N + n0 + bc);
  };

  int buf = 0;
  issue(0, 0);
  wait_async0();
  __syncthreads();
  for (int kc = 0; kc < K; kc += 16) {
    if (kc + 16 < K) issue(kc + 16, buf ^ 1);   // prefetch next chunk while computing
    #pragma unroll
    for (int kk = 0; kk < 16; kk += 4) {
      v2f a, b0, b1;
      a.x  = As[buf][mt + ml][kk + krow];
      a.y  = As[buf][mt + ml][kk + krow + 1];
      b0.x = Bs[buf][kk + krow][nt + ml];
      b0.y = Bs[buf][kk + krow + 1][nt + ml];
      b1.x = Bs[buf][kk + krow][nt + 16 + ml];
      b1.y = Bs[buf][kk + krow + 1][nt + 16 + ml];
      c0 = __builtin_amdgcn_wmma_f32_16x16x4_f32(false, a, false, b0, (short)0, c0, false, false);
      c1 = __builtin_amdgcn_wmma_f32_16x16x4_f32(false, a, false, b1, (short)0, c1, false, false);
    }
    wait_async0();       // my prefetch for buf^1 has landed
    __syncthreads();     // everyone done reading buf, all prefetches visible
    buf ^= 1;
  }
  int rhi = (lane >> 4) * 8;
  #pragma unroll
  for (int r = 0; r < 8; r++) {
    int row = m0 + mt + rhi + r;
    int col0 = n0 + nt + ml;
    int col1 = col0 + 16;
    float v0 = c0[r], v1 = c1[r];
    if (bias) { v0 += bias[col0]; v1 += bias[col1]; }
    if (res)  { v0 += res[(size_t)row * N + col0]; v1 += res[(size_t)row * N + col1]; }
    C[(size_t)row * N + col0] = v0;
    C[(size_t)row * N + col1] = v1;
  }
}

// ============ WMMA GEGLU GEMM: out[M,N] = (A@W[:,0:N]+b) * gelu(A@W[:,N:2N]+b) ============
__global__ __launch_bounds__(256) void k_gemm_geglu(const float* __restrict__ A,
                                                    const float* __restrict__ W,   // [K][2N]
                                                    const float* __restrict__ bias,// [2N]
                                                    float* __restrict__ C,         // [M][N]
                                                    int M, int N, int K) {
  __shared__ __align__(16) float As[2][64][20];
  __shared__ __align__(16) float Bs[2][16][68];
  __shared__ __align__(16) float Gs[2][16][68];
  int ntiles = N >> 6;
  int m0 = (blockIdx.x / ntiles) * 64;
  int n0 = (blockIdx.x % ntiles) * 64;
  int W2 = 2 * N;
  int tid = threadIdx.x, wave = tid >> 5, lane = tid & 31;
  int mt = (wave >> 1) * 16;
  int nt = (wave & 1) * 32;
  int krow = (lane >> 4) << 1;
  int ml = lane & 15;
  int ar = tid >> 2, ac = (tid & 3) * 4;
  int br = tid >> 4, bc = (tid & 15) * 4;
  v8f c0 = {0.f,0.f,0.f,0.f,0.f,0.f,0.f,0.f};
  v8f c1 = {0.f,0.f,0.f,0.f,0.f,0.f,0.f,0.f};
  v8f g0 = {0.f,0.f,0.f,0.f,0.f,0.f,0.f,0.f};
  v8f g1 = {0.f,0.f,0.f,0.f,0.f,0.f,0.f,0.f};

  auto issue = [&](int kc, int buf) {
    async_b128(lds_off_of(&As[buf][ar][ac]), A + (size_t)(m0 + ar) * K + kc + ac);
    async_b128(lds_off_of(&Bs[buf][br][bc]), W + (size_t)(kc + br) * W2 + n0 + bc);
    async_b128(lds_off_of(&Gs[buf][br][bc]), W + (size_t)(kc + br) * W2 + N + n0 + bc);
  };

  int buf = 0;
  issue(0, 0);
  wait_async0();
  __syncthreads();
  for (int kc = 0; kc < K; kc += 16) {
    if (kc + 16 < K) issue(kc + 16, buf ^ 1);
    #pragma unroll
    for (int kk = 0; kk < 16; kk += 4) {
      v2f a, b0, b1, h0, h1;
      a.x  = As[buf][mt + ml][kk + krow];
      a.y  = As[buf][mt + ml][kk + krow + 1];
      b0.x = Bs[buf][kk + krow][nt + ml];        b0.y = Bs[buf][kk + krow + 1][nt + ml];
      b1.x = Bs[buf][kk + krow][nt + 16 + ml];   b1.y = Bs[buf][kk + krow + 1][nt + 16 + ml];
      h0.x = Gs[buf][kk + krow][nt + ml];        h0.y = Gs[buf][kk + krow + 1][nt + ml];
      h1.x = Gs[buf][kk + krow][nt + 16 + ml];   h1.y = Gs[buf][kk + krow + 1][nt + 16 + ml];
      c0 = __builtin_amdgcn_wmma_f32_16x16x4_f32(false, a, false, b0, (short)0, c0, false, false);
      c1 = __builtin_amdgcn_wmma_f32_16x16x4_f32(false, a, false, b1, (short)0, c1, false, false);
      g0 = __builtin_amdgcn_wmma_f32_16x16x4_f32(false, a, false, h0, (short)0, g0, false, false);
      g1 = __builtin_amdgcn_wmma_f32_16x16x4_f32(false, a, false, h1, (short)0, g1, false, false);
    }
    wait_async0();
    __syncthreads();
    buf ^= 1;
  }
  int rhi = (lane >> 4) * 8;
  #pragma unroll
  for (int r = 0; r < 8; r++) {
    int row = m0 + mt + rhi + r;
    int col0 = n0 + nt + ml;
    int col1 = col0 + 16;
    float av0 = c0[r] + bias[col0];
    float av1 = c1[r] + bias[col1];
    float gv0 = g0[r] + bias[N + col0];
    float gv1 = g1[r] + bias[N + col1];
    gv0 = 0.5f * gv0 * (1.0f + erff(gv0 * 0.70710678118654752f));
    gv1 = 0.5f * gv1 * (1.0f + erff(gv1 * 0.70710678118654752f));
    C[(size_t)row * N + col0] = av0 * gv0;
    C[(size_t)row * N + col1] = av1 * gv1;
  }
}

// ---------------- split KV [T,128] -> K[T,64] (rotated), V[T,64] (opt rotated) ----------------
__global__ __launch_bounds__(256) void k_split_kv(const float* __restrict__ kv,
                                                  const float* __restrict__ cosT,
                                                  const float* __restrict__ sinT,
                                                  float* __restrict__ Ko,
                                                  float* __restrict__ Vo,
                                                  int T, int rotateV) {
  int i = blockIdx.x * 256 + threadIdx.x;
  if (i >= T * 32) return;
  int t = i >> 5, d = i & 31;
  float c = cosT[t * HD + d], s = sinT[t * HD + d];
  float k1 = kv[(size_t)t * 128 + d], k2 = kv[(size_t)t * 128 + d + 32];
  Ko[(size_t)t * HD + d]      = k1 * c - k2 * s;
  Ko[(size_t)t * HD + d + 32] = k2 * c + k1 * s;
  float v1 = kv[(size_t)t * 128 + 64 + d], v2 = kv[(size_t)t * 128 + 96 + d];
  if (rotateV) {
    Vo[(size_t)t * HD + d]      = v1 * c - v2 * s;
    Vo[(size_t)t * HD + d + 32] = v2 * c + v1 * s;
  } else {
    Vo[(size_t)t * HD + d] = v1; Vo[(size_t)t * HD + d + 32] = v2;
  }
}

// ---------------- in-place rotary over [T][ld] with H heads at col h*64 ----------------
__global__ __launch_bounds__(256) void k_rope(float* __restrict__ X,
                                              const float* __restrict__ cosT,
                                              const float* __restrict__ sinT,
                                              int T, int H, int ld, float sign) {
  int i = blockIdx.x * 256 + threadIdx.x;
  if (i >= T * H * 32) return;
  int d = i & 31;
  int h = (i >> 5) % H;
  int t = i / (32 * H);
  float c = cosT[t * HD + d], s = sinT[t * HD + d] * sign;
  float* p = X + (size_t)t * ld + h * HD;
  float x1 = p[d], x2 = p[d + 32];
  p[d]      = x1 * c - x2 * s;
  p[d + 32] = x2 * c + x1 * s;
}

// ====== ragged block-diagonal cross attention (flash style, wave32, one key/lane) ======
__global__ __launch_bounds__(256) void k_cross_attn(const float* __restrict__ Q,
                                                    const float* __restrict__ K,
                                                    const float* __restrict__ V,
                                                    const int* __restrict__ qoff,
                                                    const int* __restrict__ koff,
                                                    float* __restrict__ O, float scale) {
  __shared__ float Qs[64][HD + 1];
  __shared__ float Ks[32][HD + 1];
  __shared__ float Vs[32][HD + 1];
  int b = blockIdx.x;
  int qs = qoff[b], qe = qoff[b + 1];
  int ks = koff[b], ke = koff[b + 1];
  int tid = threadIdx.x, wave = tid >> 5, lane = tid & 31;
  for (int q0 = qs; q0 < qe; q0 += 64) {
    __syncthreads();
    for (int idx = tid; idx < 64 * HD; idx += 256) {
      int r = idx >> 6, d = idx & 63;
      int qr = q0 + r;
      Qs[r][d] = (qr < qe) ? Q[(size_t)qr * HD + d] : 0.f;
    }
    __syncthreads();
    float m[8], l[8], a0[8], a1[8];
    #pragma unroll
    for (int i = 0; i < 8; i++) { m[i] = -1e30f; l[i] = 0.f; a0[i] = 0.f; a1[i] = 0.f; }
    for (int k0 = ks; k0 < ke; k0 += 32) {
      __syncthreads();
      for (int idx = tid; idx < 32 * HD; idx += 256) {
        int r = idx >> 6, d = idx & 63;
        int kr = k0 + r;
        if (kr < ke) { Ks[r][d] = K[(size_t)kr * HD + d]; Vs[r][d] = V[(size_t)kr * HD + d]; }
        else         { Ks[r][d] = 0.f; Vs[r][d] = 0.f; }
      }
      __syncthreads();
      int nk = min(32, ke - k0);
      #pragma unroll
      for (int i = 0; i < 8; i++) {
        int q = wave * 8 + i;
        float s = 0.f;
        #pragma unroll 8
        for (int d = 0; d < HD; d++) s += Qs[q][d] * Ks[lane][d];
        s *= scale;
        if (lane >= nk) s = -1e30f;
        float sm = wred_max(s);
        float mn = fmaxf(m[i], sm);
        float p = (lane < nk) ? expf(s - mn) : 0.f;
        float ps = wred_sum(p);
        float al = expf(m[i] - mn);
        l[i] = l[i] * al + ps;
        float x0 = a0[i] * al, x1 = a1[i] * al;
        #pragma unroll 8
        for (int j = 0; j < 32; j++) {
          float pj = __shfl(p, j, 32);
          x0 += pj * Vs[j][lane];
          x1 += pj * Vs[j][lane + 32];
        }
        a0[i] = x0; a1[i] = x1; m[i] = mn;
      }
    }
    #pragma unroll
    for (int i = 0; i < 8; i++) {
      int qr = q0 + wave * 8 + i;
      if (qr < qe) {
        float inv = 1.f / l[i];
        O[(size_t)qr * HD + lane]      = a0[i] * inv;
        O[(size_t)qr * HD + lane + 32] = a1[i] * inv;
      }
    }
  }
}

// ====== self attention over [B, NL] per (batch, head); QKV packed [BNL][1536] ======
__global__ __launch_bounds__(256) void k_self_attn(const float* __restrict__ QKV,
                                                   float* __restrict__ O, float scale) {
  __shared__ float Qs[64][65], Ks[64][65], Vs[64][65];
  int blk = blockIdx.x;
  int b = blk >> 3, h = blk & 7;
  int T0 = b * NL;
  int tid = threadIdx.x, wave = tid >> 5, lane = tid & 31;
  for (int idx = tid; idx < 64 * 64; idx += 256) {
    int r = idx >> 6, d = idx & 63;
    const float* base = QKV + (size_t)(T0 + r) * 1536 + h * HD + d;
    Qs[r][d] = base[0];
    Ks[r][d] = base[512];
    Vs[r][d] = base[1024];
  }
  __syncthreads();
  float m[8], l[8], a0[8], a1[8];
  #pragma unroll
  for (int i = 0; i < 8; i++) { m[i] = -1e30f; l[i] = 0.f; a0[i] = 0.f; a1[i] = 0.f; }
  #pragma unroll
  for (int c = 0; c < 2; c++) {
    int kb = c * 32;
    #pragma unroll
    for (int i = 0; i < 8; i++) {
      int q = wave * 8 + i;
      float s = 0.f;
      #pragma unroll 8
      for (int d = 0; d < HD; d++) s += Qs[q][d] * Ks[kb + lane][d];
      s *= scale;
      float sm = wred_max(s);
      float mn = fmaxf(m[i], sm);
      float p = expf(s - mn);
      float ps = wred_sum(p);
      float al = expf(m[i] - mn);
      l[i] = l[i] * al + ps;
      float x0 = a0[i] * al, x1 = a1[i] * al;
      #pragma unroll 8
      for (int j = 0; j < 32; j++) {
        float pj = __shfl(p, j, 32);
        x0 += pj * Vs[kb + j][lane];
        x1 += pj * Vs[kb + j][lane + 32];
      }
      a0[i] = x0; a1[i] = x1; m[i] = mn;
    }
  }
  #pragma unroll
  for (int i = 0; i < 8; i++) {
    int q = wave * 8 + i;
    float inv = 1.f / l[i];
    float* o = O + (size_t)(T0 + q) * DIMM + h * HD;
    o[lane]      = a0[i] * inv;
    o[lane + 32] = a1[i] * inv;
  }
}

// ---------------- readout: out[t,0:2]=x@w0+b0, out[t,2:4]=x@w1+b1 ----------------
__global__ __launch_bounds__(256) void k_readout(const float* __restrict__ X,
                                                 const float* __restrict__ w0,
                                                 const float* __restrict__ b0,
                                                 const float* __restrict__ w1,
                                                 const float* __restrict__ b1,
                                                 float* __restrict__ out, int T) {
  int row = blockIdx.x * 8 + (threadIdx.x >> 5);
  if (row >= T) return;
  int lane = threadIdx.x & 31;
  float s00 = 0.f, s01 = 0.f, s10 = 0.f, s11 = 0.f;
  const float* x = X + (size_t)row * DIMM;
  for (int d = lane; d < DIMM; d += 32) {
    float v = x[d];
    s00 += v * w0[d * 2 + 0]; s01 += v * w0[d * 2 + 1];
    s10 += v * w1[d * 2 + 0]; s11 += v * w1[d * 2 + 1];
  }
  s00 = wred_sum(s00); s01 = wred_sum(s01);
  s10 = wred_sum(s10); s11 = wred_sum(s11);
  if (lane == 0) {
    out[(size_t)row * 4 + 0] = s00 + b0[0];
    out[(size_t)row * 4 + 1] = s01 + b0[1];
    out[(size_t)row * 4 + 2] = s10 + b1[0];
    out[(size_t)row * 4 + 3] = s11 + b1[1];
  }
}

// ---------------- param leaf indices (jax pytree: alphabetical, nested) ----------------
enum {
  DEC_BO = 0, DEC_LNC_B, DEC_LNC_G, DEC_LNQ_B, DEC_LNQ_G, DEC_WKV, DEC_WO, DEC_WQ,
  DFF_B1, DFF_B2, DFF_LN_B, DFF_LN_G, DFF_W1, DFF_W2,
  ENC_BO, ENC_LNC_B, ENC_LNC_G, ENC_LNQ_B, ENC_LNQ_G, ENC_WKV, ENC_WO, ENC_WQ,
  EFF_B1, EFF_B2, EFF_LN_B, EFF_LN_G, EFF_W1, EFF_W2,
  LAT_EMB,
  P0A_BO, P0A_LN_B, P0A_LN_G, P0A_WO, P0A_WQKV,
  P0F_B1, P0F_B2, P0F_LN_B, P0F_LN_G, P0F_W1, P0F_W2,
  P1A_BO, P1A_LN_B, P1A_LN_G, P1A_WO, P1A_WQKV,
  P1F_B1, P1F_B2, P1F_LN_B, P1F_LN_G, P1F_W1, P1F_W2,
  R_B0, R_B1, R_W0, R_W1,
  SESS_EMB, TASK_EMB, TOK_EMB, UNIT_EMB
};

extern "C" void kernel_launch(void* const* d_in, const int* in_sizes, int n_in,
                              void* d_out, int out_size, void* d_ws, size_t ws_size,
                              hipStream_t stream) {
  (void)in_sizes; (void)n_in; (void)out_size; (void)ws_size;
  const int*   uidx  = (const int*)d_in[0];
  const float* ts_in = (const float*)d_in[1];
  const int*   ttype = (const int*)d_in[2];
  const int*   lidx  = (const int*)d_in[3];
  const float* ts_lat= (const float*)d_in[4];
  const int*   sidx  = (const int*)d_in[5];
  const float* ts_out= (const float*)d_in[6];
  const int*   didx  = (const int*)d_in[7];
  const int*   ib    = (const int*)d_in[8];
  const int*   ob    = (const int*)d_in[10];
  auto P = [&](int i) -> const float* { return (const float*)d_in[11 + i]; };
  float* outp = (float*)d_out;

  // ----- workspace arena (floats) -----
  float* w = (float*)d_ws;
  int* offs = (int*)d_ws;
  size_t o = 64;
  float* cos_in = w + o; o += (size_t)TIN * HD;
  float* sin_in = w + o; o += (size_t)TIN * HD;
  float* cosL   = w + o; o += (size_t)BNL * HD;
  float* sinL   = w + o; o += (size_t)BNL * HD;
  float* cosO   = w + o; o += (size_t)TOUT * HD;
  float* sinO   = w + o; o += (size_t)TOUT * HD;
  float* Kin    = w + o; o += (size_t)TIN * HD;
  float* Vin    = w + o; o += (size_t)TIN * HD;
  float* Xn     = w + o; o += (size_t)TIN * DIMM;   // LN'd inputs (enc.lnc)
  float* hid    = w + o; o += (size_t)TOUT * FF;    // reused: kv-tmp / geglu hidden
  float* lat    = w + o; o += (size_t)BNL * DIMM;
  float* tmpA   = w + o; o += (size_t)BNL * DIMM;
  float* qry    = w + o; o += (size_t)TOUT * DIMM;
  float* tmpB   = w + o; o += (size_t)TOUT * DIMM;
  float* Qb     = w + o; o += (size_t)TOUT * HD;
  float* Ob     = w + o; o += (size_t)TOUT * HD;
  float* qkv    = w + o; o += (size_t)BNL * 1536;
  float* Osf    = w + o; o += (size_t)BNL * DIMM;
  float* Klat   = w + o; o += (size_t)BNL * HD;
  float* Vlat   = w + o; o += (size_t)BNL * HD;
  const int* in_off  = offs;
  const int* out_off = offs + 17;
  const int* lat_off = offs + 34;

  const float SCALE = 0.125f; // 1/sqrt(HD)

  auto gemm = [&](const float* A, const float* W, const float* bias, const float* res,
                  float* C, int M, int N, int K) {
    k_gemm<<<dim3((M / 64) * (N / 64)), dim3(256), 0, stream>>>(A, W, bias, res, C, M, N, K);
  };
  auto ffn = [&](float* x, int T, const float* lng, const float* lnb,
                 const float* w1, const float* b1, const float* w2, const float* b2,
                 float* lnout) {
    k_ln<<<dim3(T / 8), dim3(256), 0, stream>>>(x, lng, lnb, lnout, T);
    k_gemm_geglu<<<dim3((T / 64) * (FF / 64)), dim3(256), 0, stream>>>(lnout, w1, b1, hid, T, FF, DIMM);
    gemm(hid, w2, b2, x, x, T, DIMM, FF);
  };

  // ----- phase 0: ragged ranges + rotary tables -----
  k_offsets<<<dim3(1), dim3(256), 0, stream>>>(ib, ob, offs);
  k_rope_tables<<<dim3(TIN * 32 / 256), dim3(256), 0, stream>>>(ts_in, cos_in, sin_in, TIN);
  k_rope_tables<<<dim3(BNL * 32 / 256), dim3(256), 0, stream>>>(ts_lat, cosL, sinL, BNL);
  k_rope_tables<<<dim3(TOUT * 32 / 256), dim3(256), 0, stream>>>(ts_out, cosO, sinO, TOUT);

  // ----- encoder: latents cross-attend to inputs (rotate_value=True) -----
  k_embed_ln<<<dim3(TIN / 8), dim3(256), 0, stream>>>(uidx, ttype, P(UNIT_EMB), P(TOK_EMB),
                                                      P(ENC_LNC_G), P(ENC_LNC_B), Xn, TIN);
  gemm(Xn, P(ENC_WKV), nullptr, nullptr, hid, TIN, 2 * HD, DIMM);            // KV proj
  k_split_kv<<<dim3(TIN * 32 / 256), dim3(256), 0, stream>>>(hid, cos_in, sin_in, Kin, Vin, TIN, 1);
  k_embed1<<<dim3(BNL * DIMM / 256), dim3(256), 0, stream>>>(lidx, P(LAT_EMB), lat, BNL * DIMM);
  k_ln<<<dim3(BNL / 8), dim3(256), 0, stream>>>(lat, P(ENC_LNQ_G), P(ENC_LNQ_B), tmpA, BNL);
  gemm(tmpA, P(ENC_WQ), nullptr, nullptr, Qb, BNL, HD, DIMM);                // Q proj
  k_rope<<<dim3(BNL * 32 / 256), dim3(256), 0, stream>>>(Qb, cosL, sinL, BNL, 1, HD, 1.f);
  k_cross_attn<<<dim3(BB), dim3(256), 0, stream>>>(Qb, Kin, Vin, lat_off, in_off, Ob, SCALE);
  k_rope<<<dim3(BNL * 32 / 256), dim3(256), 0, stream>>>(Ob, cosL, sinL, BNL, 1, HD, -1.f);
  gemm(Ob, P(ENC_WO), P(ENC_BO), lat, lat, BNL, DIMM, HD);                   // out proj + residual
  ffn(lat, BNL, P(EFF_LN_G), P(EFF_LN_B), P(EFF_W1), P(EFF_B1), P(EFF_W2), P(EFF_B2), tmpA);

  // ----- process: 2 self-attention layers -----
  const float* AP[2][5] = {
    {P(P0A_BO), P(P0A_LN_B), P(P0A_LN_G), P(P0A_WO), P(P0A_WQKV)},
    {P(P1A_BO), P(P1A_LN_B), P(P1A_LN_G), P(P1A_WO), P(P1A_WQKV)}};
  const float* FP[2][6] = {
    {P(P0F_B1), P(P0F_B2), P(P0F_LN_B), P(P0F_LN_G), P(P0F_W1), P(P0F_W2)},
    {P(P1F_B1), P(P1F_B2), P(P1F_LN_B), P(P1F_LN_G), P(P1F_W1), P(P1F_W2)}};
  for (int L = 0; L < 2; L++) {
    k_ln<<<dim3(BNL / 8), dim3(256), 0, stream>>>(lat, AP[L][2], AP[L][1], tmpA, BNL);
    gemm(tmpA, AP[L][4], nullptr, nullptr, qkv, BNL, 3 * DIMM, DIMM);        // QKV proj
    // q,k,v all rotate by th_lat: 24 "heads" of 64 across the 1536-wide row
    k_rope<<<dim3(BNL * 24 * 32 / 256), dim3(256), 0, stream>>>(qkv, cosL, sinL, BNL, 24, 1536, 1.f);
    k_self_attn<<<dim3(BB * SH), dim3(256), 0, stream>>>(qkv, Osf, SCALE);
    k_rope<<<dim3(BNL * 8 * 32 / 256), dim3(256), 0, stream>>>(Osf, cosL, sinL, BNL, 8, DIMM, -1.f);
    gemm(Osf, AP[L][3], AP[L][0], lat, lat, BNL, DIMM, DIMM);                // out proj + residual
    ffn(lat, BNL, FP[L][3], FP[L][2], FP[L][4], FP[L][0], FP[L][5], FP[L][1], tmpA);
  }

  // ----- decoder: output queries cross-attend to latents (rotate_value=False) -----
  k_embed2<<<dim3(TOUT * DIMM / 256), dim3(256), 0, stream>>>(sidx, didx, P(SESS_EMB), P(TASK_EMB),
                                                              qry, TOUT * DIMM);
  k_ln<<<dim3(BNL / 8), dim3(256), 0, stream>>>(lat, P(DEC_LNC_G), P(DEC_LNC_B), tmpA, BNL);
  gemm(tmpA, P(DEC_WKV), nullptr, nullptr, hid, BNL, 2 * HD, DIMM);          // latent KV proj
  k_split_kv<<<dim3(BNL * 32 / 256), dim3(256), 0, stream>>>(hid, cosL, sinL, Klat, Vlat, BNL, 0);
  k_ln<<<dim3(TOUT / 8), dim3(256), 0, stream>>>(qry, P(DEC_LNQ_G), P(DEC_LNQ_B), tmpB, TOUT);
  gemm(tmpB, P(DEC_WQ), nullptr, nullptr, Qb, TOUT, HD, DIMM);               // query proj
  k_rope<<<dim3(TOUT * 32 / 256), dim3(256), 0, stream>>>(Qb, cosO, sinO, TOUT, 1, HD, 1.f);
  k_cross_attn<<<dim3(BB), dim3(256), 0, stream>>>(Qb, Klat, Vlat, out_off, lat_off, Ob, SCALE);
  gemm(Ob, P(DEC_WO), P(DEC_BO), qry, qry, TOUT, DIMM, HD);                  // out proj + residual
  ffn(qry, TOUT, P(DFF_LN_G), P(DFF_LN_B), P(DFF_W1), P(DFF_B1), P(DFF_W2), P(DFF_B2), tmpB);

  // ----- readout -----
  k_readout<<<dim3(TOUT / 8), dim3(256), 0, stream>>>(qry, P(R_W0), P(R_B0), P(R_W1), P(R_B1),
                                                      outp, TOUT);
}